// E2EQA_47854525612035
// MI455X (gfx1250) — compile-verified
//
#include <hip/hip_runtime.h>
#include <hip/hip_bf16.h>

#define N_R   256
#define N_E   200000
#define B_    32
#define D_Q   768
#define D_IN  1280

typedef __attribute__((ext_vector_type(2))) float v2f;
typedef __attribute__((ext_vector_type(8))) float v8f;

// ---------------------------------------------------------------------------
// Generic (32, nc) -> (nc, 32) transpose, coalesced both ways via LDS tile.
// block (32,8), grid ceil(nc/32)
// ---------------------------------------------------------------------------
__global__ void e2eqa_transpose32(const float* __restrict__ in,
                                  float* __restrict__ out, int nc) {
  __shared__ float tile[32][33];
  const int c0 = blockIdx.x * 32;
  const int tx = threadIdx.x;        // 0..31
  const int ty = threadIdx.y;        // 0..7
#pragma unroll
  for (int i = 0; i < 4; ++i) {
    int b = ty + 8 * i;              // batch row
    int c = c0 + tx;
    tile[b][tx] = (c < nc) ? in[(size_t)b * nc + c] : 0.0f;
  }
  __syncthreads();
#pragma unroll
  for (int i = 0; i < 4; ++i) {
    int cr = ty + 8 * i;             // column within tile
    int c = c0 + cr;
    if (c < nc) out[(size_t)c * 32 + tx] = tile[tx][cr];
  }
}

// ---------------------------------------------------------------------------
// Zero-fill
// ---------------------------------------------------------------------------
__global__ void e2eqa_zero(float* __restrict__ p, int n) {
  int i = blockIdx.x * blockDim.x + threadIdx.x;
  int stride = gridDim.x * blockDim.x;
  for (; i < n; i += stride) p[i] = 0.0f;
}

// ---------------------------------------------------------------------------
// Dense hop via V_WMMA_F32_16X16X4_F32:
//   r = concat @ W_inf.T   (32x1280 * 1280x256)
//   c = concat @ W_att.T
// concat = [h_q (32,768) | r1row (32,256) | r2row (32,256)] all ROW-major, so
// each WMMA A operand (two consecutive K per lane) is one aligned b64 load.
// Region loops are straight-line (no per-k branching).
// grid 1, block 512 (16 waves); wave n owns relation columns [16n, 16n+16).
// Writes r both transposed (scatter kernel) and row-major (next hop's A).
// ---------------------------------------------------------------------------
__global__ void e2eqa_dense_hop(const float* __restrict__ h_q,    // (32,768)
                                const float* __restrict__ W_inf,  // (256,1280)
                                const float* __restrict__ W_att,  // (1280)
                                const float* __restrict__ r1row,  // (32,256)|null
                                const float* __restrict__ r2row,  // (32,256)|null
                                float* __restrict__ rOutT,        // (256,32)
                                float* __restrict__ rOutRow,      // (32,256)
                                float* __restrict__ cOut,         // (32)
                                int nr1, int nr2) {               // 0 or 256
  const int lane = threadIdx.x & 31;
  const int wave = threadIdx.x >> 5;   // 0..15 -> N-tile
  const int half = lane >> 4;          // 0: lanes 0-15, 1: lanes 16-31
  const int l16  = lane & 15;

  // B operand: column j = wave*16 + l16 ; lane supplies W[j][ka], W[j][ka+1]
  const float* __restrict__ wrow = W_inf + (size_t)(wave * 16 + l16) * D_IN + half * 2;

  v8f acc0 = {};   // M-tile 0 (batch rows 0..15)
  v8f acc1 = {};   // M-tile 1 (batch rows 16..31)

  // region 0: h_q
  {
    const float* __restrict__ a0p = h_q + (size_t)l16 * D_Q + half * 2;
    const float* __restrict__ a1p = h_q + (size_t)(16 + l16) * D_Q + half * 2;
    for (int k0 = 0; k0 < D_Q; k0 += 4) {
      v2f b  = *(const v2f*)(wrow + k0);
      v2f a0 = *(const v2f*)(a0p + k0);
      v2f a1 = *(const v2f*)(a1p + k0);
      acc0 = __builtin_amdgcn_wmma_f32_16x16x4_f32(false, a0, false, b,
                                                   (short)0, acc0, false, false);
      acc1 = __builtin_amdgcn_wmma_f32_16x16x4_f32(false, a1, false, b,
                                                   (short)0, acc1, false, false);
    }
  }
  // region 1: most recent r
  if (nr1) {
    const float* __restrict__ w1  = wrow + D_Q;
    const float* __restrict__ a0p = r1row + (size_t)l16 * N_R + half * 2;
    const float* __restrict__ a1p = r1row + (size_t)(16 + l16) * N_R + half * 2;
    for (int k0 = 0; k0 < N_R; k0 += 4) {
      v2f b  = *(const v2f*)(w1 + k0);
      v2f a0 = *(const v2f*)(a0p + k0);
      v2f a1 = *(const v2f*)(a1p + k0);
      acc0 = __builtin_amdgcn_wmma_f32_16x16x4_f32(false, a0, false, b,
                                                   (short)0, acc0, false, false);
      acc1 = __builtin_amdgcn_wmma_f32_16x16x4_f32(false, a1, false, b,
                                                   (short)0, acc1, false, false);
    }
  }
  // region 2: older r
  if (nr2) {
    const float* __restrict__ w2  = wrow + D_Q + N_R;
    const float* __restrict__ a0p = r2row + (size_t)l16 * N_R + half * 2;
    const float* __restrict__ a1p = r2row + (size_t)(16 + l16) * N_R + half * 2;
    for (int k0 = 0; k0 < N_R; k0 += 4) {
      v2f b  = *(const v2f*)(w2 + k0);
      v2f a0 = *(const v2f*)(a0p + k0);
      v2f a1 = *(const v2f*)(a1p + k0);
      acc0 = __builtin_amdgcn_wmma_f32_16x16x4_f32(false, a0, false, b,
                                                   (short)0, acc0, false, false);
      acc1 = __builtin_amdgcn_wmma_f32_16x16x4_f32(false, a1, false, b,
                                                   (short)0, acc1, false, false);
    }
  }

  // D layout (16x16 f32): VGPR v holds row v (lanes 0-15) / row v+8 (lanes 16-31),
  // column = lane & 15.
  const int rel = wave * 16 + l16;
#pragma unroll
  for (int v = 0; v < 8; ++v) {
    const int m = v + half * 8;
    rOutT[rel * 32 + m]            = acc0[v];
    rOutT[rel * 32 + 16 + m]       = acc1[v];
    rOutRow[m * N_R + rel]         = acc0[v];
    rOutRow[(16 + m) * N_R + rel]  = acc1[v];
  }

  // attention scalar c[b]: wave 0, lane = batch (W_att reads are wave-uniform)
  if (wave == 0) {
    float c = 0.0f;
    const float* __restrict__ hrow = h_q + (size_t)lane * D_Q;
    for (int k = 0; k < D_Q; ++k) c += hrow[k] * W_att[k];
    if (nr1) {
      const float* __restrict__ r1r = r1row + (size_t)lane * N_R;
      for (int k = 0; k < N_R; ++k) c += r1r[k] * W_att[D_Q + k];
    }
    if (nr2) {
      const float* __restrict__ r2r = r2row + (size_t)lane * N_R;
      for (int k = 0; k < N_R; ++k) c += r2r[k] * W_att[D_Q + N_R + k];
    }
    cOut[lane] = c;
  }
}

// ---------------------------------------------------------------------------
// Sparse propagation: for each triple t:
//   yT[obj*32 + b] += xT[subj*32 + b] * r[rel*32 + b]    (b = lane)
// Indices are wave-uniform (readfirstlane) -> scalar loads; x gather and the
// atomic scatter are 128B-contiguous per triple. r table cached in LDS (32KB).
// ---------------------------------------------------------------------------
__global__ void e2eqa_hop_scatter(const float* __restrict__ xT,
                                  float* __restrict__ yT,
                                  const float* __restrict__ rT,
                                  const int* __restrict__ subj,
                                  const int* __restrict__ rel,
                                  const int* __restrict__ obj,
                                  int nT) {
  __shared__ float rlds[N_R * 32];
  for (int i = threadIdx.x; i < N_R * 32; i += blockDim.x) rlds[i] = rT[i];
  __syncthreads();

  const int lane   = threadIdx.x & 31;
  const int wave   = threadIdx.x >> 5;
  const int wpb    = blockDim.x >> 5;
  const int gwave  = blockIdx.x * wpb + wave;
  const int nWaves = gridDim.x * wpb;

  const int chunk = (nT + nWaves - 1) / nWaves;
  const int tbeg  = gwave * chunk;
  const int tend  = (tbeg + chunk < nT) ? (tbeg + chunk) : nT;

  for (int t = tbeg; t < tend; ++t) {
    const int tu = __builtin_amdgcn_readfirstlane(t);
    if ((tu & 63) == 0) {                 // stream the index arrays ahead
      __builtin_prefetch(subj + tu + 256, 0, 1);
      __builtin_prefetch(rel  + tu + 256, 0, 1);
      __builtin_prefetch(obj  + tu + 256, 0, 1);
    }
    const int s  = subj[tu];
    const int rl = rel[tu];
    const int o  = obj[tu];
    const float xv = xT[(size_t)s * 32 + lane];
    if (xv != 0.0f) {
      const float prod = xv * rlds[rl * 32 + lane];
      if (prod != 0.0f)
        atomicAdd(&yT[(size_t)o * 32 + lane], prod);
    }
  }
}

// ---------------------------------------------------------------------------
// Final combine: per-batch softmax over (c0,c1,c2), weighted sum of the three
// hop states, and transpose (N_E,32) -> (32,N_E) for the output.
// block (32,8), grid ceil(N_E/32)
// ---------------------------------------------------------------------------
__global__ void e2eqa_combine(const float* __restrict__ x1T,
                              const float* __restrict__ x2T,
                              const float* __restrict__ x3T,
                              const float* __restrict__ cbuf,  // (3,32)
                              float* __restrict__ out, int nE) {
  __shared__ float tile[32][33];
  const int e0 = blockIdx.x * 32;
  const int tx = threadIdx.x;        // batch in load phase
  const int ty = threadIdx.y;

  const float c0 = cbuf[0 * 32 + tx];
  const float c1 = cbuf[1 * 32 + tx];
  const float c2 = cbuf[2 * 32 + tx];
  const float mx = fmaxf(c0, fmaxf(c1, c2));
  const float w0 = __expf(c0 - mx);
  const float w1 = __expf(c1 - mx);
  const float w2 = __expf(c2 - mx);
  const float inv = 1.0f / (w0 + w1 + w2);
  const float a0 = w0 * inv, a1 = w1 * inv, a2 = w2 * inv;

#pragma unroll
  for (int i = 0; i < 4; ++i) {
    const int er = ty + 8 * i;
    const int e = e0 + er;
    float v = 0.0f;
    if (e < nE) {
      const size_t idx = (size_t)e * 32 + tx;
      v = a0 * x1T[idx] + a1 * x2T[idx] + a2 * x3T[idx];
    }
    tile[er][tx] = v;
  }
  __syncthreads();
#pragma unroll
  for (int i = 0; i < 4; ++i) {
    const int b = ty + 8 * i;
    const int e = e0 + tx;
    if (e < nE) out[(size_t)b * nE + e] = tile[tx][b];
  }
}

// ---------------------------------------------------------------------------
extern "C" void kernel_launch(void* const* d_in, const int* in_sizes, int n_in,
                              void* d_out, int out_size, void* d_ws, size_t ws_size,
                              hipStream_t stream) {
  const float* x     = (const float*)d_in[0];   // (32, N_E)
  const float* h_q   = (const float*)d_in[1];   // (32, 768)
  const float* W_inf = (const float*)d_in[2];   // (256, 1280)
  const float* W_att = (const float*)d_in[3];   // (1280)
  const int*   subj  = (const int*)d_in[4];
  const int*   rel   = (const int*)d_in[5];
  const int*   obj   = (const int*)d_in[6];
  const int nT = in_sizes[4];
  float* out = (float*)d_out;

  // workspace layout (floats)
  float* ws    = (float*)d_ws;
  float* rT0   = ws;                      // (256,32) transposed, 8192 each
  float* rT1   = rT0 + N_R * 32;
  float* rT2   = rT1 + N_R * 32;
  float* rRow0 = rT2 + N_R * 32;          // (32,256) row-major, 8192 each
  float* rRow1 = rRow0 + 32 * N_R;
  float* rRow2 = rRow1 + 32 * N_R;
  float* cbuf  = rRow2 + 32 * N_R;        // 3*32
  float* bufA  = ws + 65536;              // (N_E,32) = 6.4M floats each
  float* bufB  = bufA + (size_t)N_E * 32;
  float* bufC  = bufB + (size_t)N_E * 32;

  const dim3 tb(32, 8, 1);
  const int teGrid = (N_E + 31) / 32;     // 6250
  const int scatterBlocks = 2048;
  const int zeroBlocks = 1024, zeroThreads = 256;
  const int nState = N_E * 32;

  // transpose x -> bufA (x0T)
  e2eqa_transpose32<<<teGrid, tb, 0, stream>>>(x, bufA, N_E);

  // hop 0: concat = [h_q | 0 | 0] ; scatter bufA -> bufB
  e2eqa_zero<<<zeroBlocks, zeroThreads, 0, stream>>>(bufB, nState);
  e2eqa_dense_hop<<<1, 512, 0, stream>>>(h_q, W_inf, W_att, nullptr, nullptr,
                                         rT0, rRow0, cbuf + 0, 0, 0);
  e2eqa_hop_scatter<<<scatterBlocks, 256, 0, stream>>>(bufA, bufB, rT0,
                                                       subj, rel, obj, nT);

  // hop 1: concat = [h_q | r0 | 0] ; scatter bufB -> bufC
  e2eqa_zero<<<zeroBlocks, zeroThreads, 0, stream>>>(bufC, nState);
  e2eqa_dense_hop<<<1, 512, 0, stream>>>(h_q, W_inf, W_att, rRow0, nullptr,
                                         rT1, rRow1, cbuf + 32, N_R, 0);
  e2eqa_hop_scatter<<<scatterBlocks, 256, 0, stream>>>(bufB, bufC, rT1,
                                                       subj, rel, obj, nT);

  // hop 2: concat = [h_q | r1 | r0] ; scatter bufC -> bufA (x0 dead)
  e2eqa_zero<<<zeroBlocks, zeroThreads, 0, stream>>>(bufA, nState);
  e2eqa_dense_hop<<<1, 512, 0, stream>>>(h_q, W_inf, W_att, rRow1, rRow0,
                                         rT2, rRow2, cbuf + 64, N_R, N_R);
  e2eqa_hop_scatter<<<scatterBlocks, 256, 0, stream>>>(bufC, bufA, rT2,
                                                       subj, rel, obj, nT);

  // softmax-weighted combine + transpose out: y = a0*x1 + a1*x2 + a2*x3
  e2eqa_combine<<<teGrid, tb, 0, stream>>>(bufB, bufC, bufA, cbuf, out, N_E);
}